// CPC_41738492183102
// MI455X (gfx1250) — compile-verified
//
#include <hip/hip_runtime.h>
#include <hip/hip_bf16.h>
#include <math.h>

// ---------------------------------------------------------------------------
// CPC loss on gfx1250 (MI455X), fp32 WMMA (V_WMMA_F32_16X16X4_F32) everywhere.
// Sizes: B=64, ROWS=COLS=7, D=1280, H=256, S=5, NEG=16, GRU_ROWS=6
// ---------------------------------------------------------------------------

typedef __attribute__((ext_vector_type(2))) float v2f;
typedef __attribute__((ext_vector_type(8))) float v8f;

__device__ __forceinline__ v8f wmma_f32(v2f a, v2f b, v8f c) {
  // D = A(16x4) * B(4x16) + C(16x16), all fp32
  return __builtin_amdgcn_wmma_f32_16x16x4_f32(
      /*neg_a=*/false, a, /*neg_b=*/false, b,
      /*c_mod=*/(short)0, c, /*reuse_a=*/false, /*reuse_b=*/false);
}

// ---------------------------------------------------------------------------
// Kernel 1: gi[m, j] = sum_d X[m,d] * W_ih[j,d] + b_ih[j]
//   m = t*448 + (b*7+c)  (M=2688), j in [0,768), K=1280
//   X row (t,b,c) == encodings[b, t, c, :]
//   Wave tile: 16M x 64N (A frag reused over 4 N-subtiles). Block: 4 waves.
// ---------------------------------------------------------------------------
__global__ __launch_bounds__(128) void cpc_gi_kernel(
    const float* __restrict__ enc, const float* __restrict__ wih,
    const float* __restrict__ bih, float* __restrict__ gi) {
  const int lane = threadIdx.x & 31;
  const int wave = threadIdx.x >> 5;
  const int hl = lane & 15;
  const int kq = (lane >> 4) * 2;         // 0 or 2: K sub-offset per half-wave
  const int m0 = blockIdx.x * 16;
  const int n0 = blockIdx.y * 256 + wave * 64;

  // Map A row -> encodings address
  const int m = m0 + hl;                   // 0..2687
  const int t = m / 448;
  const int n = m - t * 448;               // 0..447
  const int bb = n / 7, cc = n - bb * 7;
  const float* aptr = enc + (size_t)(((bb * 7 + t) * 7 + cc)) * 1280 + kq;

  const float* bptr[4];
#pragma unroll
  for (int j = 0; j < 4; ++j)
    bptr[j] = wih + (size_t)(n0 + j * 16 + hl) * 1280 + kq;

  v8f acc[4] = {};
  for (int k = 0; k < 1280; k += 4) {
    v2f a = *(const v2f*)(aptr + k);
#pragma unroll
    for (int j = 0; j < 4; ++j) {
      v2f w = *(const v2f*)(bptr[j] + k);
      acc[j] = wmma_f32(a, w, acc[j]);
    }
  }

  const int row_base = m0 + 8 * (lane >> 4);
#pragma unroll
  for (int j = 0; j < 4; ++j) {
    const int col = n0 + j * 16 + hl;
    const float bias = bih[col];
#pragma unroll
    for (int v = 0; v < 8; ++v)
      gi[(size_t)(row_base + v) * 768 + col] = acc[j][v] + bias;
  }
}

// ---------------------------------------------------------------------------
// Kernel 2: GRU recurrence over t=0..5 for 16 sequences per block.
//   gh = h * W_hh^T  (16x768, K=256), gates elementwise, h update in LDS.
//   Block: 8 waves; wave w owns gh columns [w*96, w*96+96).
//   ctx[(b*6+t)*7+c, :] = h_new
// ---------------------------------------------------------------------------
#define H_LD 260   // 256 + 4 pad (stride % 64 dwords == 4 -> conflict-free frags)
#define GH_LD 772  // 768 + 4 pad

__global__ __launch_bounds__(256) void cpc_gru_kernel(
    const float* __restrict__ gi, const float* __restrict__ whh,
    const float* __restrict__ bhh, const float* __restrict__ hidden,
    float* __restrict__ ctx) {
  __shared__ float sh[16 * H_LD];
  __shared__ float sgh[16 * GH_LD];
  const int tid = threadIdx.x;
  const int lane = tid & 31, wave = tid >> 5;
  const int hl = lane & 15, kq = (lane >> 4) * 2;
  const int n0 = blockIdx.x * 16;

  // h0 = broadcast(hidden)
  for (int idx = tid; idx < 16 * 256; idx += 256) {
    const int seq = idx >> 8, j = idx & 255;
    sh[seq * H_LD + j] = hidden[j];
  }
  __syncthreads();

  const int nb0 = wave * 96;
  const float* bp[6];
#pragma unroll
  for (int jt = 0; jt < 6; ++jt)
    bp[jt] = whh + (size_t)(nb0 + jt * 16 + hl) * 256 + kq;

  const float* ap = sh + hl * H_LD + kq;

  for (int t = 0; t < 6; ++t) {
    v8f acc[6] = {};
    for (int k = 0; k < 256; k += 4) {
      v2f a = *(const v2f*)(ap + k);
#pragma unroll
      for (int jt = 0; jt < 6; ++jt) {
        v2f w = *(const v2f*)(bp[jt] + k);
        acc[jt] = wmma_f32(a, w, acc[jt]);
      }
    }
    // spill gh tile to LDS
    const int rb = 8 * (lane >> 4);
#pragma unroll
    for (int jt = 0; jt < 6; ++jt) {
      const int col = nb0 + jt * 16 + hl;
#pragma unroll
      for (int v = 0; v < 8; ++v)
        sgh[(rb + v) * GH_LD + col] = acc[jt][v];
    }
    __syncthreads();

    // gates: each thread handles 16 (seq, j) pairs
    for (int idx = tid; idx < 16 * 256; idx += 256) {
      const int seq = idx >> 8, j = idx & 255;
      const int ng = n0 + seq;
      const float* gir = gi + (size_t)(t * 448 + ng) * 768;
      const float i_r = gir[j];
      const float i_z = gir[j + 256];
      const float i_n = gir[j + 512];
      const float h_r = sgh[seq * GH_LD + j] + bhh[j];
      const float h_z = sgh[seq * GH_LD + j + 256] + bhh[j + 256];
      const float h_n = sgh[seq * GH_LD + j + 512] + bhh[j + 512];
      const float r = 1.0f / (1.0f + expf(-(i_r + h_r)));
      const float z = 1.0f / (1.0f + expf(-(i_z + h_z)));
      const float nn = tanhf(i_n + r * h_n);
      const float ho = sh[seq * H_LD + j];
      const float hn = (1.0f - z) * nn + z * ho;
      sh[seq * H_LD + j] = hn;
      const int bb = ng / 7, cc = ng - bb * 7;
      ctx[(size_t)((bb * 6 + t) * 7 + cc) * 256 + j] = hn;
    }
    __syncthreads();
  }
}

// ---------------------------------------------------------------------------
// Kernel 3: per (b, s) block.
//   GEMM1: pred[rc, d] = clip(ctx[b]·wk_w[s]^T + wk_b[s]), rc=42 (pad 48), K=256
//          -> staged into LDS in 5 chunks of 256 d-columns.
//   GEMM2: dots[rc, q] += pred_chunk · enc[b]^T (q=49 pad 64), acc in registers.
//   Epilogue: gather pos + 16 negs, log-softmax(17), masked sums -> atomics.
// ---------------------------------------------------------------------------
#define P_LD 260
#define DOT_LD 68

__global__ __launch_bounds__(256) void cpc_dots_kernel(
    const float* __restrict__ ctx, const float* __restrict__ enc,
    const float* __restrict__ wkw, const float* __restrict__ wkb,
    const int* __restrict__ neg_rows, const int* __restrict__ neg_cols,
    float* __restrict__ accum) {
  __shared__ float spred[48 * P_LD];
  __shared__ float sdots[48 * DOT_LD];
  __shared__ float sred[2];
  const int tid = threadIdx.x, lane = tid & 31, wave = tid >> 5;
  const int hl = lane & 15, kq = (lane >> 4) * 2;
  const int b = blockIdx.x, s = blockIdx.y;

  // GEMM1 A (ctx rows, clamped at 41 for the pad rows)
  const float* aP[3];
#pragma unroll
  for (int mi = 0; mi < 3; ++mi) {
    int m = mi * 16 + hl;
    if (m > 41) m = 41;
    aP[mi] = ctx + (size_t)(b * 42 + m) * 256 + kq;
  }

  // GEMM2 tile ownership: 12 tiles (mi=id/4, nj=id%4); waves 0-3 own {w, w+8}
  const int tcount = (wave < 4) ? 2 : 1;
  int d_mi[2], d_nj[2];
  const float* q_enc[2];
  v8f dacc[2] = {};
  for (int u = 0; u < tcount; ++u) {
    const int id = (u == 0) ? wave : wave + 8;
    d_mi[u] = id >> 2;
    d_nj[u] = id & 3;
    int q = d_nj[u] * 16 + hl;
    if (q > 48) q = 48;  // pad columns read a valid row; results unused
    q_enc[u] = enc + (size_t)(b * 49 + q) * 1280 + kq;
  }

  for (int ch = 0; ch < 5; ++ch) {
    const int d0 = ch * 256;

    // ---- GEMM1: 3 M-tiles x 2 N-subtiles per wave (nj = 2*wave + njl) ----
    v8f pacc[3][2] = {};
    const float* bP[2];
#pragma unroll
    for (int njl = 0; njl < 2; ++njl) {
      const int d = d0 + (wave * 2 + njl) * 16 + hl;
      bP[njl] = wkw + ((size_t)s * 1280 + d) * 256 + kq;
    }
    for (int k = 0; k < 256; k += 4) {
      v2f a0 = *(const v2f*)(aP[0] + k);
      v2f a1 = *(const v2f*)(aP[1] + k);
      v2f a2 = *(const v2f*)(aP[2] + k);
      v2f w0 = *(const v2f*)(bP[0] + k);
      v2f w1 = *(const v2f*)(bP[1] + k);
      pacc[0][0] = wmma_f32(a0, w0, pacc[0][0]);
      pacc[1][0] = wmma_f32(a1, w0, pacc[1][0]);
      pacc[2][0] = wmma_f32(a2, w0, pacc[2][0]);
      pacc[0][1] = wmma_f32(a0, w1, pacc[0][1]);
      pacc[1][1] = wmma_f32(a1, w1, pacc[1][1]);
      pacc[2][1] = wmma_f32(a2, w1, pacc[2][1]);
    }

    __syncthreads();  // previous chunk's GEMM2 finished reading spred

    // bias + hardtanh, spill pred chunk to LDS
#pragma unroll
    for (int njl = 0; njl < 2; ++njl) {
      const int colL = (wave * 2 + njl) * 16 + hl;  // 0..255 within chunk
      const float bias = wkb[(size_t)s * 1280 + d0 + colL];
#pragma unroll
      for (int mi = 0; mi < 3; ++mi) {
        const int rb = mi * 16 + 8 * (lane >> 4);
#pragma unroll
        for (int v = 0; v < 8; ++v) {
          float p = pacc[mi][njl][v] + bias;
          p = fminf(1.0f, fmaxf(-1.0f, p));
          spred[(rb + v) * P_LD + colL] = p;
        }
      }
    }
    __syncthreads();

    // ---- GEMM2: dots += pred_chunk(48x256) · enc_chunk^T (64) ----
    for (int u = 0; u < tcount; ++u) {
      const float* apl = spred + (d_mi[u] * 16 + hl) * P_LD + kq;
      const float* bpe = q_enc[u] + d0;
      v8f acc = dacc[u];
      for (int k = 0; k < 256; k += 4) {
        v2f a = *(const v2f*)(apl + k);
        v2f e = *(const v2f*)(bpe + k);
        acc = wmma_f32(a, e, acc);
      }
      dacc[u] = acc;
    }
  }

  // spill dots to LDS
  for (int u = 0; u < tcount; ++u) {
    const int rb = d_mi[u] * 16 + 8 * (lane >> 4);
    const int col = d_nj[u] * 16 + hl;
#pragma unroll
    for (int v = 0; v < 8; ++v)
      sdots[(rb + v) * DOT_LD + col] = dacc[u][v];
  }
  if (tid < 2) sred[tid] = 0.0f;
  __syncthreads();

  // ---- epilogue: log-softmax over [pos, 16 negs], masked by s + r < 6 ----
  if (tid < 42) {
    const int r = tid / 7, c = tid - r * 7;
    if (s + r < 6) {
      float d[17];
      d[0] = sdots[tid * DOT_LD + (r * 7 + c)];
      const int base = (((b * 5 + s) * 6 + r) * 7 + c) * 16;
#pragma unroll
      for (int e = 0; e < 16; ++e) {
        const int q = neg_rows[base + e] * 7 + neg_cols[base + e];
        d[e + 1] = sdots[tid * DOT_LD + q];
      }
      float mx = d[0];
      bool ok = true;  // argmax == 0 (ties resolve to index 0)
#pragma unroll
      for (int e = 1; e < 17; ++e) {
        if (d[e] > mx) mx = d[e];
        if (d[e] > d[0]) ok = false;
      }
      float se = 0.0f;
#pragma unroll
      for (int e = 0; e < 17; ++e) se += expf(d[e] - mx);
      const float loss = (logf(se) + mx) - d[0];
      atomicAdd(&sred[0], loss);
      atomicAdd(&sred[1], ok ? 1.0f : 0.0f);
    }
  }
  __syncthreads();
  if (tid == 0) {
    atomicAdd(&accum[0], sred[0]);
    atomicAdd(&accum[1], sred[1]);
  }
}

// ---------------------------------------------------------------------------
__global__ void cpc_init_kernel(float* __restrict__ accum) {
  if (threadIdx.x < 2) accum[threadIdx.x] = 0.0f;
}

__global__ void cpc_final_kernel(const float* __restrict__ accum,
                                 float* __restrict__ out) {
  if (threadIdx.x == 0) {
    const float inv = 1.0f / 8960.0f;  // 20 masked rows * 64 * 7
    out[0] = accum[0] * inv;
    out[1] = accum[1] * inv;
  }
}

// ---------------------------------------------------------------------------
extern "C" void kernel_launch(void* const* d_in, const int* in_sizes, int n_in,
                              void* d_out, int out_size, void* d_ws,
                              size_t ws_size, hipStream_t stream) {
  const float* enc    = (const float*)d_in[0];  // [64,7,7,1280]
  const float* hidden = (const float*)d_in[1];  // [1,1,256]
  const float* wih    = (const float*)d_in[2];  // [768,1280]
  const float* whh    = (const float*)d_in[3];  // [768,256]
  const float* bih    = (const float*)d_in[4];  // [768]
  const float* bhh    = (const float*)d_in[5];  // [768]
  const float* wkw    = (const float*)d_in[6];  // [5,1280,256]
  const float* wkb    = (const float*)d_in[7];  // [5,1280]
  const int*   nr     = (const int*)d_in[8];    // [64,5,6,7,16]
  const int*   nc     = (const int*)d_in[9];    // [64,5,6,7,16]
  float* out = (float*)d_out;

  float* wsf   = (float*)d_ws;
  float* accum = wsf;                    // 2 floats (+pad)
  float* gi    = wsf + 16;               // 2688 x 768
  float* ctx   = gi + 2688 * 768;        // 2688 x 256 == [B,6,7,256]

  cpc_init_kernel<<<1, 32, 0, stream>>>(accum);
  cpc_gi_kernel<<<dim3(168, 3), 128, 0, stream>>>(enc, wih, bih, gi);
  cpc_gru_kernel<<<28, 256, 0, stream>>>(gi, whh, bhh, hidden, ctx);
  cpc_dots_kernel<<<dim3(64, 5), 256, 0, stream>>>(ctx, enc, wkw, wkb, nr, nc,
                                                   accum);
  cpc_final_kernel<<<1, 32, 0, stream>>>(accum, out);
}